// GatedGATLayer_55576876810744
// MI455X (gfx1250) — compile-verified
//
#include <hip/hip_runtime.h>
#include <hip/hip_bf16.h>

// GatedGAT layer for MI455X (gfx1250, wave32).
// B=2, N=512, H=64.  Outputs: X_next (2*512*64) then E_out (2*512*512*64), f32.
// Fused design: per (b,i) block stages E[b,i,:,:] (128KB) into LDS once via the
// Tensor Data Mover (padded to a 68-float row stride for bank-conflict-free
// WMMA fragment reads), computes softmax logits from LDS, then the 512x64x64
// GEMM via v_wmma_f32_16x16x4_f32, LayerNorm in-register, stores E_out.

#define BB 2
#define NN 512
#define HH 64
#define NROW (BB * NN)       // 1024 (b,i) rows
#define EPSLN 1e-5f
#define EPITCH 68            // padded LDS row stride (floats)

typedef __attribute__((ext_vector_type(2))) float v2f;
typedef __attribute__((ext_vector_type(8))) float v8f;
typedef __attribute__((ext_vector_type(4))) unsigned int v4u;
typedef __attribute__((ext_vector_type(8))) int v8i;
typedef __attribute__((ext_vector_type(4))) int v4i;

// LDS floats: Etile 512*68 + bfr 4096 + attRow 512 + agg 64 + red 8
#define LDS_FLOATS (NN * EPITCH + 4096 + NN + HH + 8)

// ---------------------------------------------------------------- kernel 0
// column sums of W3/W4:  c[h] = sum_o W[o][h]
__global__ void gat_colsum_k(const float* __restrict__ W3,
                             const float* __restrict__ W4,
                             float* __restrict__ c3, float* __restrict__ c4) {
  int h = threadIdx.x;            // 64 threads
  float s3 = 0.f, s4 = 0.f;
  for (int o = 0; o < HH; ++o) { s3 += W3[o * HH + h]; s4 += W4[o * HH + h]; }
  c3[h] = s3; c4[h] = s4;
}

// ---------------------------------------------------------------- kernel 1
// per X-row: xi_s = X.c3 ;  Xj = X@W5^T ;  Xres = X@W6^T
__global__ __launch_bounds__(64)
void gat_xprep_k(const float* __restrict__ X, const float* __restrict__ W5,
                 const float* __restrict__ W6, const float* __restrict__ c3,
                 float* __restrict__ xi_s, float* __restrict__ Xj,
                 float* __restrict__ Xres) {
  int r = blockIdx.x;             // 0..1023
  int t = threadIdx.x;            // output channel o
  __shared__ float xs[HH];
  __shared__ float red[HH];
  xs[t] = X[r * HH + t];
  __syncthreads();
  red[t] = xs[t] * c3[t];
  __syncthreads();
  for (int s = 32; s > 0; s >>= 1) { if (t < s) red[t] += red[t + s]; __syncthreads(); }
  if (t == 0) xi_s[r] = red[0];
  float a5 = 0.f, a6 = 0.f;
  for (int h = 0; h < HH; ++h) {
    float xv = xs[h];
    a5 += xv * W5[t * HH + h];
    a6 += xv * W6[t * HH + h];
  }
  Xj[r * HH + t]   = a5;
  Xres[r * HH + t] = a6;
}

// ---------------------------------------------------------------- kernel 2
// Fused: TDM-stage E row-block -> logits+softmax -> WMMA GEMM -> att scale
// -> LayerNorm -> E_out store + agg_msg reduction.
__global__ __launch_bounds__(256)
void gat_fused_k(const float* __restrict__ E, const float* __restrict__ W4,
                 const float* __restrict__ c4, const float* __restrict__ xi_s,
                 const float* __restrict__ Xj,
                 const float* __restrict__ ge_g, const float* __restrict__ ge_b,
                 float* __restrict__ Eout, float* __restrict__ agg) {
  extern __shared__ float lds[];
  float* Etile  = lds;                    // 512 * 68
  float* bfr    = lds + NN * EPITCH;      // 4096 (W4^T fragments)
  float* attRow = bfr + 4096;             // 512
  float* aggLds = attRow + NN;            // 64
  float* redS   = aggLds + HH;            // 8

  int r = blockIdx.x;             // b*512 + i
  int b = r >> 9;
  int t = threadIdx.x;
  int lane = t & 31, wave = t >> 5;
  const float* Erow = E + (size_t)r * NN * HH;
  float* EoutRow = Eout + (size_t)r * NN * HH;

  // ---- B-matrix (W4^T, 64x64) pre-swizzled into per-lane WMMA frag layout:
  // frag f = kk*4+nn, lane l holds {K=kbase, K=kbase+1} of column o.
  for (int s = t; s < 2048; s += 256) {
    int f = s >> 5, l = s & 31;
    int kk = f >> 2, nn = f & 3;
    int k = kk * 4 + ((l < 16) ? 0 : 2);
    int o = nn * 16 + (l & 15);
    bfr[s * 2 + 0] = W4[o * HH + k];       // B[k][o]   = W4[o][k]
    bfr[s * 2 + 1] = W4[o * HH + k + 1];   // B[k+1][o]
  }
  if (t < HH) aggLds[t] = 0.f;

  // ---- stage E[b,i,:,:] (512x64 f32) into LDS with 68-float row pitch ----
#if defined(__has_builtin) && __has_builtin(__builtin_amdgcn_tensor_load_to_lds)
  if (wave == 0) {
    unsigned long long ga = (unsigned long long)(uintptr_t)Erow;
    unsigned int ldsoff = (unsigned int)(uintptr_t)Etile; // low 32 bits = LDS byte addr
    v4u g0;
    g0.x = 1u;                                           // count=1, user descriptor
    g0.y = ldsoff;                                       // lds_addr
    g0.z = (unsigned int)(ga & 0xffffffffull);           // global_addr[31:0]
    g0.w = (unsigned int)((ga >> 32) & 0x1ffffffull)     // global_addr[56:32]
         | (2u << 30);                                   // type=2 (image)
    v8i g1;
    // data_size=4B (2<<16) | pad_enable (1<<20) | pad_interval=64 DW (5<<22)
    // | pad_amount=4 DW (3<<25)  -> LDS pitch 68 floats
    g1[0] = (int)((2u << 16) | (1u << 20) | (5u << 22) | (3u << 25));
    g1[1] = (int)(64u  << 16);   // tensor_dim0 = 64
    g1[2] = (int)(512u << 16);   // tensor_dim1 = 512
    g1[3] = (int)(64u  << 16);   // tile_dim0 = 64
    g1[4] = 512;                 // tile_dim1 = 512
    g1[5] = 64;                  // tensor_dim0_stride = 64
    g1[6] = 0;
    g1[7] = 0;
    v4i gz4 = (v4i){0, 0, 0, 0};
    v8i gz8 = (v8i){0, 0, 0, 0, 0, 0, 0, 0};
    // clang-23 / therock-10.0 6-arg form:
    // (v4u g0, v8i g1, v4i g2, v4i g3, v8i extra, i32 cpol)
    __builtin_amdgcn_tensor_load_to_lds(g0, g1, gz4, gz4, gz8, 0);
    __builtin_amdgcn_s_wait_tensorcnt(0);
  }
#else
  for (int idx = t; idx < NN * 16; idx += 256) {
    int j = idx >> 4, seg = idx & 15;
    float4 v = *reinterpret_cast<const float4*>(Erow + j * HH + seg * 4);
    *reinterpret_cast<float4*>(&Etile[j * EPITCH + seg * 4]) = v;
  }
#endif
  __syncthreads();

  // ---- logits: attRow[j] = xi_s[i] + xi_s[j] + dot(E[j,:], c4) ----
  float cc0 = c4[lane * 2], cc1 = c4[lane * 2 + 1];
  float xii = xi_s[r];
  for (int k = 0; k < 64; ++k) {
    int j = wave + 8 * k;
    v2f ev = *reinterpret_cast<const v2f*>(&Etile[j * EPITCH + lane * 2]);
    float d = ev.x * cc0 + ev.y * cc1;
    for (int off = 16; off > 0; off >>= 1) d += __shfl_down(d, off, 32);
    if (lane == 0) attRow[j] = d + xii + xi_s[(b << 9) + j];
  }
  __syncthreads();

  // ---- softmax over attRow[512] ----
  float l0 = attRow[t], l1 = attRow[t + 256];
  float m = fmaxf(l0, l1);
  for (int off = 16; off > 0; off >>= 1) m = fmaxf(m, __shfl_xor(m, off, 32));
  if (lane == 0) redS[wave] = m;
  __syncthreads();
  if (t == 0) { float mm = redS[0]; for (int w = 1; w < 8; ++w) mm = fmaxf(mm, redS[w]); redS[0] = mm; }
  __syncthreads();
  float M = redS[0];
  __syncthreads();
  float e0 = __expf(l0 - M), e1 = __expf(l1 - M);
  float s = e0 + e1;
  for (int off = 16; off > 0; off >>= 1) s += __shfl_xor(s, off, 32);
  if (lane == 0) redS[wave] = s;
  __syncthreads();
  if (t == 0) { float ss = 0.f; for (int w = 0; w < 8; ++w) ss += redS[w]; redS[0] = ss; }
  __syncthreads();
  float inv = 1.f / redS[0];
  attRow[t]       = e0 * inv;
  attRow[t + 256] = e1 * inv;
  __syncthreads();

  // ---- GEMM (E @ W4^T) + att scale + LayerNorm + agg ----
  float gg[4], gb[4];
  for (int nn = 0; nn < 4; ++nn) {
    int o = nn * 16 + (lane & 15);
    gg[nn] = ge_g[o]; gb[nn] = ge_b[o];
  }
  const v8f vzero = (v8f){0.f, 0.f, 0.f, 0.f, 0.f, 0.f, 0.f, 0.f};
  float accAgg[4] = {0.f, 0.f, 0.f, 0.f};
  int mrow  = lane & 15;
  int mbase = (lane < 16) ? 0 : 8;

  for (int tile = wave; tile < 32; tile += 8) {   // uniform per wave: EXEC all ones
    int j0 = tile * 16;
    v8f c[4] = {vzero, vzero, vzero, vzero};
    for (int kk = 0; kk < 16; ++kk) {
      // A frag (16x4 f32): lanes 0-15 K={4kk,4kk+1}; lanes 16-31 K={4kk+2,4kk+3}
      int kbase = kk * 4 + ((lane < 16) ? 0 : 2);
      v2f a = *reinterpret_cast<const v2f*>(&Etile[(j0 + mrow) * EPITCH + kbase]);
      for (int nn = 0; nn < 4; ++nn) {
        v2f bf = *reinterpret_cast<const v2f*>(&bfr[((kk * 4 + nn) * 32 + lane) * 2]);
        c[nn] = __builtin_amdgcn_wmma_f32_16x16x4_f32(
            false, a, false, bf, (short)0, c[nn], false, false);
      }
    }

    // scale by att[j]; per-row sum / sum-of-squares for LayerNorm
    float av[8], sum[8], ssum[8];
    for (int rr = 0; rr < 8; ++rr) av[rr] = attRow[j0 + rr + mbase];
    for (int rr = 0; rr < 8; ++rr) {
      float sv = 0.f, ssv = 0.f;
      for (int nn = 0; nn < 4; ++nn) {
        float v = c[nn][rr] * av[rr];
        c[nn][rr] = v;
        sv += v; ssv += v * v;
      }
      sum[rr] = sv; ssum[rr] = ssv;
    }
    for (int off = 1; off < 16; off <<= 1) {   // reduce across 16-lane halves
      for (int rr = 0; rr < 8; ++rr) {
        sum[rr]  += __shfl_xor(sum[rr],  off, 32);
        ssum[rr] += __shfl_xor(ssum[rr], off, 32);
      }
    }
    for (int rr = 0; rr < 8; ++rr) {
      float mean = sum[rr] * (1.f / 64.f);
      float var  = ssum[rr] * (1.f / 64.f) - mean * mean;
      float rs   = rsqrtf(var + EPSLN);
      int j = j0 + rr + mbase;
      for (int nn = 0; nn < 4; ++nn) {
        float v = c[nn][rr];
        int o = nn * 16 + (lane & 15);
        EoutRow[(size_t)j * HH + o] = (v - mean) * rs * gg[nn] + gb[nn];
        accAgg[nn] += v * Xj[((b << 9) + j) * HH + o];
      }
    }
  }
  for (int nn = 0; nn < 4; ++nn) {
    int h = nn * 16 + (lane & 15);
    atomicAdd(&aggLds[h], accAgg[nn]);     // ds_add_f32
  }
  __syncthreads();
  if (t < HH) agg[r * HH + t] = aggLds[t];
}

// ---------------------------------------------------------------- kernel 3
// X_next = LayerNorm(relu(agg + Xres)) + X
__global__ __launch_bounds__(64)
void gat_xfinal_k(const float* __restrict__ X, const float* __restrict__ agg,
                  const float* __restrict__ Xres, const float* __restrict__ gn_g,
                  const float* __restrict__ gn_b, float* __restrict__ Xout) {
  int r = blockIdx.x;
  int h = threadIdx.x;
  __shared__ float red[HH];
  float v = fmaxf(agg[r * HH + h] + Xres[r * HH + h], 0.f);
  red[h] = v;
  __syncthreads();
  for (int s = 32; s > 0; s >>= 1) { if (h < s) red[h] += red[h + s]; __syncthreads(); }
  float mean = red[0] * (1.f / 64.f);
  __syncthreads();
  float d = v - mean;
  red[h] = d * d;
  __syncthreads();
  for (int s = 32; s > 0; s >>= 1) { if (h < s) red[h] += red[h + s]; __syncthreads(); }
  float rs = rsqrtf(red[0] * (1.f / 64.f) + EPSLN);
  Xout[r * HH + h] = d * rs * gn_g[h] + gn_b[h] + X[r * HH + h];
}

// ---------------------------------------------------------------- launch
extern "C" void kernel_launch(void* const* d_in, const int* in_sizes, int n_in,
                              void* d_out, int out_size, void* d_ws, size_t ws_size,
                              hipStream_t stream) {
  const float* X    = (const float*)d_in[0];
  const float* E    = (const float*)d_in[1];
  const float* W3   = (const float*)d_in[2];
  const float* W4   = (const float*)d_in[3];
  const float* W5   = (const float*)d_in[4];
  const float* W6   = (const float*)d_in[5];
  const float* gn_g = (const float*)d_in[6];
  const float* gn_b = (const float*)d_in[7];
  const float* ge_g = (const float*)d_in[8];
  const float* ge_b = (const float*)d_in[9];
  (void)in_sizes; (void)n_in; (void)out_size; (void)ws_size;

  float* ws   = (float*)d_ws;
  float* c3   = ws;              // 64
  float* c4   = ws + 64;         // 64
  float* xi_s = ws + 128;        // 1024
  float* Xj   = ws + 1152;       // 65536
  float* Xres = ws + 66688;      // 65536
  float* agg  = ws + 132224;     // 65536   (total ~0.8 MB)

  float* Xout = (float*)d_out;           // 2*512*64
  float* Eout = Xout + NROW * HH;        // 2*512*512*64

  const size_t shbytes = (size_t)LDS_FLOATS * sizeof(float);   // ~158 KB
  (void)hipFuncSetAttribute(reinterpret_cast<const void*>(&gat_fused_k),
                            hipFuncAttributeMaxDynamicSharedMemorySize,
                            (int)shbytes);

  gat_colsum_k<<<dim3(1),    dim3(64),  0, stream>>>(W3, W4, c3, c4);
  gat_xprep_k <<<dim3(NROW), dim3(64),  0, stream>>>(X, W5, W6, c3, xi_s, Xj, Xres);
  gat_fused_k <<<dim3(NROW), dim3(256), shbytes, stream>>>(E, W4, c4, xi_s, Xj,
                                                           ge_g, ge_b, Eout, agg);
  gat_xfinal_k<<<dim3(NROW), dim3(64),  0, stream>>>(X, agg, Xres, gn_g, gn_b, Xout);
}